// SSIMLoss_edit_13666585936695
// MI455X (gfx1250) — compile-verified
//
#include <hip/hip_runtime.h>
#include <hip/hip_bf16.h>

typedef float v2f __attribute__((ext_vector_type(2)));
typedef float v4f __attribute__((ext_vector_type(4)));
typedef float v8f __attribute__((ext_vector_type(8)));

// Exact fp32 wave-wide (32-lane) sum using V_WMMA_F32_16X16X4_F32.
// A layout (16x4 f32): lanes 0-15 -> {VGPR0:K0, VGPR1:K1} for M=lane,
//                      lanes 16-31 -> {VGPR0:K2, VGPR1:K3} for M=lane-16.
// With a = {v, 0}: A[M][0] = p[M], A[M][2] = p[M+16], rest 0.
// With B = all ones, D[M][*] = p[M] + p[M+16].
// D layout: VGPR j holds row M=j (lanes 0-15) / M=j+8 (lanes 16-31).
// Sum of 8 D VGPRs -> rows 0-7 in lanes 0-15, rows 8-15 in lanes 16-31;
// one xor-16 shuffle completes the full 32-lane sum (all lanes get it).
// Requires EXEC == all ones (call only from fully-converged code).
__device__ __forceinline__ float wave_sum_wmma(float v) {
  v2f a; a.x = v;    a.y = 0.0f;
  v2f b; b.x = 1.0f; b.y = 1.0f;
  v8f c = {};
  v8f d = __builtin_amdgcn_wmma_f32_16x16x4_f32(
      /*neg_a=*/false, a, /*neg_b=*/false, b,
      /*c_mod=*/(short)0, c, /*reuse_a=*/false, /*reuse_b=*/false);
  float s = ((d[0] + d[1]) + (d[2] + d[3])) + ((d[4] + d[5]) + (d[6] + d[7]));
  s += __shfl_xor(s, 16, 32);
  return s;
}

// Block-wide sum (valid in thread 0). blockDim.x must be a multiple of 32.
__device__ __forceinline__ float block_sum(float v, float* lds) {
  float ws = wave_sum_wmma(v);
  const int lane = threadIdx.x & 31;
  const int wid  = threadIdx.x >> 5;
  if (lane == 0) lds[wid] = ws;
  __syncthreads();
  float total = 0.0f;
  if (threadIdx.x == 0) {
    const int nw = blockDim.x >> 5;
    for (int i = 0; i < nw; ++i) total += lds[i];
  }
  return total;
}

__device__ __forceinline__ float window_term4(v4f xv, v4f rv) {
  // Per-window (size 4): mean + std contributions, exact reference math.
  float sx = (xv.x + xv.y) + (xv.z + xv.w);
  float qx = (xv.x * xv.x + xv.y * xv.y) + (xv.z * xv.z + xv.w * xv.w);
  float sr = (rv.x + rv.y) + (rv.z + rv.w);
  float qr = (rv.x * rv.x + rv.y * rv.y) + (rv.z * rv.z + rv.w * rv.w);
  float mx = sx * 0.25f, mr = sr * 0.25f;
  float vx = fmaxf(qx * 0.25f - mx * mx, 0.0f);
  float vr = fmaxf(qr * 0.25f - mr * mr, 0.0f);
  return (mx - mr) + (sqrtf(vx) - sqrtf(vr));
}

__global__ void ssim_partial_kernel(const float* __restrict__ recon,
                                    const float* __restrict__ x,
                                    const int* __restrict__ winp,
                                    long long n,
                                    float* __restrict__ partials) {
  __shared__ float lds[32];
  const int win = *winp;                       // uniform scalar load
  const long long m = n / (long long)win;
  const long long tid    = (long long)blockIdx.x * blockDim.x + threadIdx.x;
  const long long stride = (long long)gridDim.x * blockDim.x;

  float acc = 0.0f;
  if (win == 4) {
    // One window == one 16B vector: perfect global_load_b128 coalescing.
    // Dataset (537 MB) > L2 (192 MB) and is single-use -> non-temporal.
    const v4f* __restrict__ x4 = (const v4f*)x;
    const v4f* __restrict__ r4 = (const v4f*)recon;
#pragma unroll 4
    for (long long w = tid; w < m; w += stride) {
      v4f xv = __builtin_nontemporal_load(x4 + w);
      v4f rv = __builtin_nontemporal_load(r4 + w);
      acc += window_term4(xv, rv);
    }
  } else {
    const float inv = 1.0f / (float)win;
    for (long long w = tid; w < m; w += stride) {
      const float* xp = x + w * win;
      const float* rp = recon + w * win;
      float sx = 0.f, qx = 0.f, sr = 0.f, qr = 0.f;
      for (int i = 0; i < win; ++i) {
        float xe = xp[i]; sx += xe; qx += xe * xe;
        float re = rp[i]; sr += re; qr += re * re;
      }
      float mx = sx * inv, mr = sr * inv;
      float vx = fmaxf(qx * inv - mx * mx, 0.0f);
      float vr = fmaxf(qr * inv - mr * mr, 0.0f);
      acc += (mx - mr) + (sqrtf(vx) - sqrtf(vr));
    }
  }

  float total = block_sum(acc, lds);
  if (threadIdx.x == 0) partials[blockIdx.x] = total;
}

__global__ void ssim_final_kernel(const float* __restrict__ partials,
                                  int num_partials,
                                  const int* __restrict__ winp,
                                  long long n,
                                  float* __restrict__ out) {
  __shared__ float lds[32];
  const int win = *winp;
  const long long m = n / (long long)win;

  float acc = 0.0f;
  for (int i = threadIdx.x; i < num_partials; i += blockDim.x)
    acc += partials[i];

  float total = block_sum(acc, lds);
  if (threadIdx.x == 0) out[0] = total / (float)m;
}

extern "C" void kernel_launch(void* const* d_in, const int* in_sizes, int n_in,
                              void* d_out, int out_size, void* d_ws, size_t ws_size,
                              hipStream_t stream) {
  const float* recon = (const float*)d_in[0];
  const float* x     = (const float*)d_in[1];
  const int*   winp  = (const int*)d_in[2];
  const long long n  = (long long)in_sizes[1];

  const int threads = 256;                     // 8 wave32s per block
  int blocks = 4096;                           // 1M threads; 16 windows/thread
  size_t max_blocks = ws_size / sizeof(float); // partials live in d_ws
  if ((size_t)blocks > max_blocks) blocks = (int)max_blocks;
  if (blocks < 1) blocks = 1;

  float* partials = (float*)d_ws;
  ssim_partial_kernel<<<blocks, threads, 0, stream>>>(recon, x, winp, n, partials);
  ssim_final_kernel<<<1, threads, 0, stream>>>(partials, blocks, winp, n, (float*)d_out);
}